// MDAHead_55576876810464
// MI455X (gfx1250) — compile-verified
//
#include <hip/hip_runtime.h>
#include <hip/hip_bf16.h>

// Problem constants (from the reference): B=256, C=1000, K=2, D=128.
#define Bsz   256
#define Csz   1000
#define Kmix  2
#define Dsz   128
#define CKsz  (Csz * Kmix)   // 2000
#define NBLK  (Dsz / 16)     // 8 block rows of 16
#define BT    64             // batch columns per workgroup (4 waves x 16 cols)
#define LDY   65             // padded LDS row stride (floats)

typedef __attribute__((ext_vector_type(2))) float v2f;
typedef __attribute__((ext_vector_type(8))) float v8f;

// ---------------------------------------------------------------------------
// Kernel 1: blocked forward substitution  L y = (z - mu)  per component,
// Mahalanobis accumulation, logdet, writes logpdf (without mixture weights)
// into W[ck * B + b].
//
// grid = (B/BT, CK), block = 128 threads = 4 waves (wave32).
// Wave w owns the 16 batch columns [w*16, w*16+16) of this workgroup's tile.
//
// Trick: LDS holds -Y (negated solved columns), so the off-diagonal update
//   T = RHS - sum_{j<i} L_ij * Y_j
// becomes a pure multiply-accumulate  c += L_ij * (-Y_j)  with no per-chunk
// VALU negation feeding the WMMAs. Two accumulators break the WMMA RAW chain.
// ---------------------------------------------------------------------------
__global__ __launch_bounds__(128)
void trsm_maha_kernel(const float* __restrict__ z,      // (B, D)
                      const float* __restrict__ mu,     // (CK, D)
                      const float* __restrict__ covL,   // (CK, D, D) lower-tri
                      float* __restrict__ W)            // (CK, B)
{
    __shared__ float Ysh[Dsz][LDY];   // -Y (and staged T), 33 KB
    __shared__ float sLii[16][17];    // current diagonal block
    __shared__ float sInv[16];        // 1 / diag
    __shared__ float sLogdet;         // sum log(diag) accumulator

    const int ck    = blockIdx.y;
    const int btile = blockIdx.x;
    const int tid   = threadIdx.x;
    const int lane  = tid & 31;
    const int wave  = tid >> 5;        // 0..3
    const int half  = lane >> 4;       // lane 0-15 vs 16-31
    const int l15   = lane & 15;

    const float* __restrict__ Lck  = covL + (size_t)ck * Dsz * Dsz;
    const float* __restrict__ muck = mu   + (size_t)ck * Dsz;

    const int colofs = wave * 16;           // column offset within tile
    const int n      = colofs + l15;        // this lane's column (0..63)
    const int bcol   = btile * BT + n;      // global batch index

    if (tid == 0) sLogdet = 0.0f;
    __syncthreads();

    float maha = 0.0f;                      // meaningful for tid < 64

    for (int i = 0; i < NBLK; ++i) {
        // ---- stage diagonal block L[i][i] into LDS (2 elems / thread) ----
        {
            int idx = tid;
            int r = idx >> 4, s = idx & 15;
            sLii[r][s] = Lck[(i * 16 + r) * Dsz + i * 16 + s];
            idx = tid + 128; r = idx >> 4; s = idx & 15;
            sLii[r][s] = Lck[(i * 16 + r) * Dsz + i * 16 + s];
        }
        __syncthreads();
        if (tid < 16) {
            float dg = sLii[tid][tid];
            sInv[tid] = 1.0f / dg;
            atomicAdd(&sLogdet, __logf(dg));       // ds_add_f32
        }

        // ---- init accumulators: c0 = RHS = z - mu, c1 = 0
        //      (WMMA C layout: VGPR v -> row M = v + 8*half, lane&15 = N) ----
        v8f c0, c1;
#pragma unroll
        for (int v = 0; v < 8; ++v) {
            int d = i * 16 + v + half * 8;
            c0[v] = z[bcol * Dsz + d] - muck[d];
            c1[v] = 0.0f;
        }

        // prefetch next block row of L while we compute (gfx1250 path)
        if (i + 1 < NBLK)
            __builtin_prefetch(Lck + ((i + 1) * 16 + l15) * Dsz, 0, 1);

        // ---- c += sum_{j<i} L[i][j] * (-Y[j])  via fp32 WMMA ----
        // A (16x4 f32): lane&15 = row M; VGPR0 -> K = koff, VGPR1 -> K = koff+1
        // B (4x16 f32): lane&15 = col N; VGPR0 -> K = koff, VGPR1 -> K = koff+1
        const int row  = l15;
        const int koff = half * 2;
        for (int j = 0; j < i; ++j) {
#pragma unroll
            for (int q = 0; q < 4; ++q) {
                int kk = q * 4 + koff;
                v2f a = *reinterpret_cast<const v2f*>(
                            Lck + (i * 16 + row) * Dsz + j * 16 + kk);
                v2f b;
                b.x = Ysh[j * 16 + kk    ][n];
                b.y = Ysh[j * 16 + kk + 1][n];
                if (q & 1) {
                    c1 = __builtin_amdgcn_wmma_f32_16x16x4_f32(
                            false, a, false, b, (short)0, c1, false, false);
                } else {
                    c0 = __builtin_amdgcn_wmma_f32_16x16x4_f32(
                            false, a, false, b, (short)0, c0, false, false);
                }
            }
        }

        // ---- T = c0 + c1; spill to LDS for the column-per-lane solve ----
#pragma unroll
        for (int v = 0; v < 8; ++v)
            Ysh[i * 16 + v + half * 8][n] = c0[v] + c1[v];
        __syncthreads();

        // ---- serial substitution on the 16x16 diagonal block:
        //      one batch column per thread (threads 0..63).
        //      Write back NEGATED y so later GEMMs need no negation. ----
        if (tid < 64) {
            const int col = tid;
            float y[16];
#pragma unroll
            for (int r2 = 0; r2 < 16; ++r2) y[r2] = Ysh[i * 16 + r2][col];
#pragma unroll
            for (int r2 = 0; r2 < 16; ++r2) {
                float acc = y[r2];
                for (int s2 = 0; s2 < r2; ++s2)
                    acc += sLii[r2][s2] * y[s2];   // y[] holds -y values
                acc *= sInv[r2];
                maha += acc * acc;
                y[r2] = -acc;                       // store -y
                Ysh[i * 16 + r2][col] = -acc;
            }
        }
        __syncthreads();   // -Y[i] visible for later GEMMs; sLii reusable
    }

    // ---- logpdf = -0.5 * (maha + logdet + D*log(2*pi)) ----
    if (tid < 64) {
        const float LOG2PI = 1.8378770664093454f;
        float lp = -0.5f * (maha + 2.0f * sLogdet + (float)Dsz * LOG2PI);
        W[(size_t)ck * Bsz + btile * BT + tid] = lp;
    }
}

// ---------------------------------------------------------------------------
// Kernel 2: per (b, c): log-softmax over K=2 mixture logits, logsumexp with
// logpdf, add prior. out[b*C + c].
// ---------------------------------------------------------------------------
__global__ __launch_bounds__(256)
void combine_kernel(const float* __restrict__ W,          // (CK, B)
                    const float* __restrict__ logits_pi,  // (C, K)
                    const float* __restrict__ prior,      // (C,)
                    float* __restrict__ out)              // (B, C)
{
    int idx = blockIdx.x * blockDim.x + threadIdx.x;
    if (idx >= Bsz * Csz) return;
    int b = idx / Csz;
    int cc = idx - b * Csz;

    float a0 = W[(size_t)(cc * 2 + 0) * Bsz + b];
    float a1 = W[(size_t)(cc * 2 + 1) * Bsz + b];
    float p0 = logits_pi[cc * 2 + 0];
    float p1 = logits_pi[cc * 2 + 1];

    // log_softmax over K=2
    float pm   = fmaxf(p0, p1);
    float plse = pm + __logf(__expf(p0 - pm) + __expf(p1 - pm));
    float x0 = a0 + (p0 - plse);
    float x1 = a1 + (p1 - plse);

    float m = fmaxf(x0, x1);
    float r = m + __logf(__expf(x0 - m) + __expf(x1 - m));
    out[idx] = prior[cc] + r;
}

extern "C" void kernel_launch(void* const* d_in, const int* in_sizes, int n_in,
                              void* d_out, int out_size, void* d_ws, size_t ws_size,
                              hipStream_t stream) {
    const float* z          = (const float*)d_in[0];  // (B, D)
    const float* mu         = (const float*)d_in[1];  // (C, K, D)
    const float* logits_pi  = (const float*)d_in[2];  // (C, K)
    const float* covL       = (const float*)d_in[3];  // (C, K, D, D)
    const float* prior      = (const float*)d_in[4];  // (C,)
    float*       out        = (float*)d_out;          // (B, C)
    float*       W          = (float*)d_ws;           // (CK, B) scratch

    dim3 grid1(Bsz / BT, CKsz);   // (4, 2000)
    trsm_maha_kernel<<<grid1, 128, 0, stream>>>(z, mu, covL, W);

    int n = Bsz * Csz;
    combine_kernel<<<(n + 255) / 256, 256, 0, stream>>>(W, logits_pi, prior, out);
}